// PAGTNGNN_87960930222247
// MI455X (gfx1250) — compile-verified
//
#include <hip/hip_runtime.h>
#include <hip/hip_bf16.h>

// ---------------------------------------------------------------------------
// PAGTN GNN forward on gfx1250 (CDNA5, wave32, WMMA).
// Dense projections: persistent-wave f32 WMMA GEMM (V_WMMA_F32_16X16X4_F32)
// with register-resident B fragments (loaded once per wave, reused across all
// M tiles of that wave's column block).
// Edge phase: fused edge-parallel kernels (L2-resident node-table gathers,
// wave32 butterfly reductions, f32 global atomics for segment softmax/scatter).
// ---------------------------------------------------------------------------

#define N_NODES 50000
#define N_EDGES 400000
#define F_IN    128
#define F_HID   32
#define N_HEADS 4
#define NH      (N_NODES * N_HEADS)

typedef float v2f __attribute__((ext_vector_type(2)));
typedef float v8f __attribute__((ext_vector_type(8)));

// ------------------------- persistent WMMA GEMM -----------------------------
// C[M x NC] = A[M x K] @ B[K x NC] + bias[NC], optional ReLU.
// One 16x16 C tile per WMMA issue; each wave owns a fixed N-column tile and
// strides over M tiles, so its K/4 B fragments live in VGPRs for the whole
// kernel. M % 16 == 0, K % 4 == 0, NC % 16 == 0 hold for every instance.
// All branches are wave-uniform -> EXEC stays all-ones around WMMA.
template <int K, int NC>
__global__ void wmma_gemm_persist(const float* __restrict__ A,
                                  const float* __restrict__ B,
                                  const float* __restrict__ bias,
                                  float* __restrict__ C,
                                  int M, int relu, int wavesTotal) {
    constexpr int KSTEPS  = K / 4;
    constexpr int TILES_N = NC / 16;

    const int lane = threadIdx.x & 31;
    const int wgid = blockIdx.x * (blockDim.x >> 5) + (threadIdx.x >> 5);
    const int tn   = wgid % TILES_N;           // this wave's column tile
    const int mt0  = wgid / TILES_N;           // first M tile
    const int mstr = wavesTotal / TILES_N;     // M-tile stride (exact divide)

    const int half = lane >> 4;                // selects K pair {0,1} vs {2,3}
    const int l16  = lane & 15;
    const int col  = tn * 16 + l16;            // B/C column held by this lane

    // ---- preload all B fragments + bias once (register-resident) ----
    v2f bf[KSTEPS];
#pragma unroll
    for (int s = 0; s < KSTEPS; ++s) {
        const int k = 4 * s + 2 * half;
        bf[s].x = B[(size_t)k * NC + col];
        bf[s].y = B[(size_t)(k + 1) * NC + col];
    }
    const float bv = bias[col];

    const int mtiles = M / 16;
    for (int mt = mt0; mt < mtiles; mt += mstr) {
        const float* __restrict__ Ap = A + (size_t)(mt * 16 + l16) * K + 2 * half;
        v2f af[KSTEPS];
#pragma unroll
        for (int s = 0; s < KSTEPS; ++s) {     // K/4 x b64 loads, coalesced rows
            af[s].x = Ap[4 * s];
            af[s].y = Ap[4 * s + 1];
        }
        v8f acc = {0.f, 0.f, 0.f, 0.f, 0.f, 0.f, 0.f, 0.f};
#pragma unroll
        for (int s = 0; s < KSTEPS; ++s)       // back-to-back WMMA chain
            acc = __builtin_amdgcn_wmma_f32_16x16x4_f32(
                false, af[s], false, bf[s], (short)0, acc, false, false);

        float* __restrict__ Cp = C + (size_t)(mt * 16 + 8 * half) * NC + col;
#pragma unroll
        for (int v = 0; v < 8; ++v) {          // row = mt*16 + v + 8*half
            float r = acc[v] + bv;
            if (relu) r = r > 0.f ? r : 0.f;
            Cp[(size_t)v * NC] = r;
        }
    }
}

// ------------------------- order-preserving f32 key -------------------------
__device__ __forceinline__ unsigned enc_f32(float f) {
    unsigned u = __float_as_uint(f);
    return (u & 0x80000000u) ? ~u : (u | 0x80000000u);
}
__device__ __forceinline__ float dec_f32(unsigned u) {
    return __uint_as_float((u & 0x80000000u) ? (u & 0x7FFFFFFFu) : ~u);
}

// ----------------------------- edge attention -------------------------------
// One wave per edge, lane = feature f in [0,32).
// alpha_raw[e][h] = relu(xs[dst[e]][h] + xd[src[e]][h] + ea[e]) . dw + db
__global__ void edge_alpha_kernel(const float* __restrict__ xs,
                                  const float* __restrict__ xd,
                                  const float* __restrict__ ea,
                                  const int* __restrict__ src,
                                  const int* __restrict__ dst,
                                  const float* __restrict__ dw,
                                  const float* __restrict__ db,
                                  float* __restrict__ alpha, int E) {
    const int lane = threadIdx.x & 31;
    const int e = blockIdx.x * (blockDim.x >> 5) + (threadIdx.x >> 5);
    if (e >= E) return;
    const int s = src[e], d = dst[e];
    const float eav = ea[(size_t)e * 32 + lane];
    const float wf  = dw[lane];

    float p0, p1, p2, p3;
    {
        float t;
        t = xs[(size_t)(d * 4 + 0) * 32 + lane] + xd[(size_t)(s * 4 + 0) * 32 + lane] + eav;
        p0 = (t > 0.f ? t : 0.f) * wf;
        t = xs[(size_t)(d * 4 + 1) * 32 + lane] + xd[(size_t)(s * 4 + 1) * 32 + lane] + eav;
        p1 = (t > 0.f ? t : 0.f) * wf;
        t = xs[(size_t)(d * 4 + 2) * 32 + lane] + xd[(size_t)(s * 4 + 2) * 32 + lane] + eav;
        p2 = (t > 0.f ? t : 0.f) * wf;
        t = xs[(size_t)(d * 4 + 3) * 32 + lane] + xd[(size_t)(s * 4 + 3) * 32 + lane] + eav;
        p3 = (t > 0.f ? t : 0.f) * wf;
    }
#pragma unroll
    for (int off = 16; off > 0; off >>= 1) {
        p0 += __shfl_xor(p0, off, 32);
        p1 += __shfl_xor(p1, off, 32);
        p2 += __shfl_xor(p2, off, 32);
        p3 += __shfl_xor(p3, off, 32);
    }
    float r = p0;
    if (lane == 1) r = p1;
    else if (lane == 2) r = p2;
    else if (lane == 3) r = p3;
    if (lane < 4) alpha[(size_t)e * 4 + lane] = r + db[0];
}

// segment max over dst (shared across heads), via monotone-uint atomicMax
__global__ void alpha_max_kernel(const float* __restrict__ alpha,
                                 const int* __restrict__ dst,
                                 unsigned* __restrict__ nmax, int E) {
    const int e = blockIdx.x * blockDim.x + threadIdx.x;
    if (e >= E) return;
    float m = alpha[(size_t)e * 4 + 0];
    m = fmaxf(m, alpha[(size_t)e * 4 + 1]);
    m = fmaxf(m, alpha[(size_t)e * 4 + 2]);
    m = fmaxf(m, alpha[(size_t)e * 4 + 3]);
    atomicMax(&nmax[dst[e]], enc_f32(m));
}

// exp(alpha - max[dst]) in place + segment sum over dst (heads folded in)
__global__ void alpha_expsum_kernel(float* __restrict__ alpha,
                                    const int* __restrict__ dst,
                                    const unsigned* __restrict__ nmax,
                                    float* __restrict__ nsum, int E) {
    const int e = blockIdx.x * blockDim.x + threadIdx.x;
    if (e >= E) return;
    const int d = dst[e];
    const float nm = dec_f32(nmax[d]);
    float s = 0.f;
#pragma unroll
    for (int h = 0; h < 4; ++h) {
        float ex = expf(alpha[(size_t)e * 4 + h] - nm);
        alpha[(size_t)e * 4 + h] = ex;
        s += ex;
    }
    atomicAdd(&nsum[d], s);
}

// -------------------------- message + scatter -------------------------------
// One wave per edge, lane = feature. Normalizes alpha on the fly and
// reproduces the reference's head-tiled index quirk:
//   message k = e*H + h  scatters to flat segment  dst[k mod E]  of (N*H, F).
__global__ void message_kernel(const float* __restrict__ xm,
                               const float* __restrict__ em,
                               const float* __restrict__ alpha,
                               const float* __restrict__ nsum,
                               const int* __restrict__ dst,
                               float* __restrict__ aggr, int E) {
    const int lane = threadIdx.x & 31;
    const int e = blockIdx.x * (blockDim.x >> 5) + (threadIdx.x >> 5);
    if (e >= E) return;
    const int d = dst[e];
    const float inv = 1.0f / (nsum[d] + 1e-8f);
    const float emv = em[(size_t)e * 32 + lane];
#pragma unroll
    for (int h = 0; h < 4; ++h) {
        const float a = alpha[(size_t)e * 4 + h] * inv;
        const float v = a * (xm[(size_t)(d * 4 + h) * 32 + lane] + emv);
        const int k = e * 4 + h;
        const int km = k - (k / E) * E;           // k mod E
        const int seg = dst[km];                  // flat (N*H) segment id
        atomicAdd(&aggr[(size_t)seg * 32 + lane], v);
    }
}

// h_new = relu(aggr + (h @ wnw + wnb) + atom_in)
__global__ void combine_kernel(const float* __restrict__ aggr,
                               const float* __restrict__ hw,
                               const float* __restrict__ atom_in,
                               float* __restrict__ hout, int n) {
    const int i = blockIdx.x * blockDim.x + threadIdx.x;
    if (i >= n) return;
    const float v = aggr[i] + hw[i] + atom_in[i];
    hout[i] = v > 0.f ? v : 0.f;
}

// out[n][f] = mean over heads of h[n][h][f]
__global__ void mean_kernel(const float* __restrict__ h,
                            float* __restrict__ out, int total) {
    const int i = blockIdx.x * blockDim.x + threadIdx.x;
    if (i >= total) return;
    const int n = i >> 5, f = i & 31;
    float s = 0.f;
#pragma unroll
    for (int hh = 0; hh < 4; ++hh) s += h[(size_t)(n * 4 + hh) * 32 + f];
    out[i] = s * 0.25f;
}

// ---------------------------------------------------------------------------
extern "C" void kernel_launch(void* const* d_in, const int* in_sizes, int n_in,
                              void* d_out, int out_size, void* d_ws, size_t ws_size,
                              hipStream_t stream) {
    const float* x         = (const float*)d_in[0];
    const float* edge_attr = (const float*)d_in[1];
    const int*   eidx      = (const int*)d_in[2];
    const float* atom_w    = (const float*)d_in[3];
    const float* atom_b    = (const float*)d_in[4];
    const float* asw = (const float*)d_in[5],  *asb = (const float*)d_in[6];
    const float* adw = (const float*)d_in[7],  *adb = (const float*)d_in[8];
    const float* aew = (const float*)d_in[9],  *aeb = (const float*)d_in[10];
    const float* dwv = (const float*)d_in[11], *dbv = (const float*)d_in[12];
    const float* mdw = (const float*)d_in[13], *mdb = (const float*)d_in[14];
    const float* mew = (const float*)d_in[15], *meb = (const float*)d_in[16];
    const float* wnw = (const float*)d_in[17], *wnb = (const float*)d_in[18];
    float* out = (float*)d_out;

    const int* src = eidx;
    const int* dst = eidx + N_EDGES;

    // ---- workspace carve (floats) ----
    float* ws = (float*)d_ws;
    size_t o = 0;
    float* atom_in = ws + o; o += (size_t)N_NODES * F_IN;   // (N,H,F) flat
    float* hbuf    = ws + o; o += (size_t)N_NODES * F_IN;
    float* xsb     = ws + o; o += (size_t)NH * F_HID;
    float* xdb     = ws + o; o += (size_t)NH * F_HID;
    float* xmb     = ws + o; o += (size_t)NH * F_HID;
    float* hwb     = ws + o; o += (size_t)NH * F_HID;
    float* eab     = ws + o; o += (size_t)N_EDGES * F_HID;
    float* emb     = ws + o; o += (size_t)N_EDGES * F_HID;
    float* alpha   = ws + o; o += (size_t)N_EDGES * N_HEADS;
    unsigned* nmax = (unsigned*)(ws + o); o += N_NODES;
    float* nsum    = ws + o; o += N_NODES;
    float* aggr    = ws + o; o += (size_t)NH * F_HID;

    // Persistent grid: 512 blocks x 4 waves = 2048 waves; divisible by
    // tilesN = 2 (NC=32) and tilesN = 8 (NC=128) so the M-stride is exact.
    constexpr int PBLOCKS = 512;
    constexpr int PWAVES  = PBLOCKS * 4;

    auto gemm128 = [&](const float* A, const float* B, const float* bias,
                       float* C, int M, int relu) {
        wmma_gemm_persist<128, 128><<<PBLOCKS, 128, 0, stream>>>(
            A, B, bias, C, M, relu, PWAVES);
    };
    auto gemm32 = [&](const float* A, const float* B, const float* bias,
                      float* C, int M, int relu) {
        wmma_gemm_persist<32, 32><<<PBLOCKS, 128, 0, stream>>>(
            A, B, bias, C, M, relu, PWAVES);
    };

    // atom_in = relu(x @ atom_w + atom_b)
    gemm128(x, atom_w, atom_b, atom_in, N_NODES, 1);

    const int FF = F_HID * F_HID;
    for (int l = 0; l < 2; ++l) {
        const float* hin = (l == 0) ? atom_in : hbuf;

        gemm32(hin, asw + l * FF, asb + l * F_HID, xsb, NH, 0);
        gemm32(hin, adw + l * FF, adb + l * F_HID, xdb, NH, 0);
        gemm32(hin, mdw + l * FF, mdb + l * F_HID, xmb, NH, 0);
        gemm32(hin, wnw + l * FF, wnb + l * F_HID, hwb, NH, 0);
        gemm32(edge_attr, aew + l * FF, aeb + l * F_HID, eab, N_EDGES, 0);
        gemm32(edge_attr, mew + l * FF, meb + l * F_HID, emb, N_EDGES, 0);

        hipMemsetAsync(nmax, 0, N_NODES * sizeof(unsigned), stream);
        hipMemsetAsync(nsum, 0, N_NODES * sizeof(float), stream);
        hipMemsetAsync(aggr, 0, (size_t)NH * F_HID * sizeof(float), stream);

        edge_alpha_kernel<<<N_EDGES / 8, 256, 0, stream>>>(
            xsb, xdb, eab, src, dst, dwv + l * F_HID, dbv + l, alpha, N_EDGES);
        alpha_max_kernel<<<(N_EDGES + 255) / 256, 256, 0, stream>>>(
            alpha, dst, nmax, N_EDGES);
        alpha_expsum_kernel<<<(N_EDGES + 255) / 256, 256, 0, stream>>>(
            alpha, dst, nmax, nsum, N_EDGES);
        message_kernel<<<N_EDGES / 8, 256, 0, stream>>>(
            xmb, emb, alpha, nsum, dst, aggr, N_EDGES);
        combine_kernel<<<(NH * F_HID + 255) / 256, 256, 0, stream>>>(
            aggr, hwb, atom_in, hbuf, NH * F_HID);
    }

    mean_kernel<<<(N_NODES * F_HID + 255) / 256, 256, 0, stream>>>(
        hbuf, out, N_NODES * F_HID);
}